// EVModel_53790170415115
// MI455X (gfx1250) — compile-verified
//
#include <hip/hip_runtime.h>

typedef __attribute__((ext_vector_type(2))) float v2f;
typedef __attribute__((ext_vector_type(4))) float v4f;
typedef __attribute__((ext_vector_type(8))) float v8f;
typedef __attribute__((ext_vector_type(4))) int   v4i;

#define N_ENT     100000
#define N_REL     250000
#define N_TRIG    50000
#define N_ARGS    250000
#define ENT_DIM   288
#define REL_R     256
#define RTYPE_DIM 32
#define ROLE_DIM  256
#define ARG_DIM   576            // 256 + 32 + 288
#define OUT_DIM   544            // 288 + 256

#define TILE_M    64             // args per block (4 WMMA sub-tiles of 16)
#define SUBT      (TILE_M / 16)  // 4
#define XS_STRIDE 580            // 576 + 4 pad -> conflict-free A-fragment reads

// ---------------------------------------------------------------------------
// Kernel 1: out[:, :288] = ent_embeds[trig_ent_id]; out[:, 288:544] = 0
// ---------------------------------------------------------------------------
__global__ __launch_bounds__(256)
void ev_init_out(const float* __restrict__ ent,
                 const int*   __restrict__ trig_ent,
                 float*       __restrict__ out)
{
    size_t idx = (size_t)blockIdx.x * blockDim.x + threadIdx.x;
    const size_t total = (size_t)N_TRIG * OUT_DIM;
    if (idx >= total) return;
    int row = (int)(idx / OUT_DIM);
    int c   = (int)(idx % OUT_DIM);
    float v = 0.0f;
    if (c < ENT_DIM) {
        v = ent[(size_t)trig_ent[row] * ENT_DIM + c];
    }
    out[idx] = v;
}

// ---------------------------------------------------------------------------
// Kernel 2: per-arg masked GEMM via V_WMMA_F32_16X16X4_F32 + segment-sum
// ---------------------------------------------------------------------------
__global__ __launch_bounds__(512)
void ev_args_wmma(const float* __restrict__ ent,
                  const float* __restrict__ rel,
                  const float* __restrict__ rtab,
                  const float* __restrict__ Win,
                  const float* __restrict__ Wout,
                  const int*   __restrict__ rtype_ids,
                  const int*   __restrict__ arg_trig,
                  const int*   __restrict__ arg_rel,
                  const int*   __restrict__ arg_ent,
                  const int*   __restrict__ arg_is_in,
                  float*       __restrict__ out,
                  float*       __restrict__ sink)
{
    extern __shared__ float smem[];
    float* xs    = smem;                              // TILE_M * XS_STRIDE
    float* smask = xs + TILE_M * XS_STRIDE;           // TILE_M
    int*   strig = (int*)(smask + TILE_M);            // TILE_M

    const int tile = blockIdx.x;
    const int t    = threadIdx.x;

    // ---- stage x = [rel | rtype | ent] rows into LDS (8 threads / row) ----
    {
        const int row = t >> 3;          // 0..63
        const int sub = t & 7;           // 0..7
        const int g   = tile * TILE_M + row;
        const bool valid = (g < N_ARGS);
        int rrel = 0, rent = 0, rt = 0;
        if (valid) {
            rrel = arg_rel[g];
            rent = arg_ent[g];
            rt   = rtype_ids[rrel];
        }
        if (sub == 0) {
            strig[row] = valid ? arg_trig[g] : -1;
            smask[row] = valid ? (float)arg_is_in[g] : 0.0f;
        }
        float* dst = xs + row * XS_STRIDE;
        #pragma unroll
        for (int k = 0; k < 18; ++k) {
            const int q = k * 8 + sub;   // float4 index 0..143
            const int j = q * 4;         // float offset within row
            v4f val = {0.0f, 0.0f, 0.0f, 0.0f};
            if (valid) {
                if (j < REL_R) {                       // k < 8
                    val = *(const v4f*)(rel + (size_t)rrel * REL_R + j);
                } else if (j < REL_R + RTYPE_DIM) {    // k == 8
                    val = *(const v4f*)(rtab + (size_t)rt * RTYPE_DIM + (j - REL_R));
                } else {                               // k >= 9
                    val = *(const v4f*)(ent + (size_t)rent * ENT_DIM + (j - REL_R - RTYPE_DIM));
                }
            }
            *(v4f*)(dst + j) = val;
        }
    }
    __syncthreads();

    // ---- compute: wave -> 16-column N tile, four 16-row M sub-tiles ----
    const int wave = t >> 5;             // 0..15
    const int lane = t & 31;
    const int l    = lane & 15;          // M (A) / N (B,C,D)
    const int h    = lane >> 4;          // K-pair selector
    const int ncol = wave * 16 + l;

    float m[SUBT];
    #pragma unroll
    for (int s = 0; s < SUBT; ++s) m[s] = smask[s * 16 + l];

    v8f acc[SUBT];
    #pragma unroll
    for (int s = 0; s < SUBT; ++s) acc[s] = (v8f){};

    #pragma unroll 2
    for (int k0 = 0; k0 < ARG_DIM; k0 += 4) {
        const int ka = k0 + 2 * h;

        v2f bin, bout;
        bin.x  = Win [ka       * ROLE_DIM + ncol];
        bin.y  = Win [(ka + 1) * ROLE_DIM + ncol];
        bout.x = Wout[ka       * ROLE_DIM + ncol];
        bout.y = Wout[(ka + 1) * ROLE_DIM + ncol];

        v2f a[SUBT], ai[SUBT], ao[SUBT];
        #pragma unroll
        for (int s = 0; s < SUBT; ++s) {
            a[s] = *(const v2f*)(xs + (size_t)(s * 16 + l) * XS_STRIDE + ka);
            ai[s].x = a[s].x * m[s];  ai[s].y = a[s].y * m[s];
            ao[s].x = a[s].x - ai[s].x;  ao[s].y = a[s].y - ai[s].y; // exact: m in {0,1}
        }
        #pragma unroll
        for (int s = 0; s < SUBT; ++s) {
            acc[s] = __builtin_amdgcn_wmma_f32_16x16x4_f32(false, ai[s], false, bin,
                                                           (short)0, acc[s], false, false);
        }
        #pragma unroll
        for (int s = 0; s < SUBT; ++s) {
            acc[s] = __builtin_amdgcn_wmma_f32_16x16x4_f32(false, ao[s], false, bout,
                                                           (short)0, acc[s], false, false);
        }
    }

    // ---- segment-sum scatter: C/D layout -> M = v + 8*h, N = l ----
    // Trigger ids hoisted to registers (2x ds_load_b128 per sub-tile); invalid
    // rows (tail block only) are redirected branchlessly into a scratch sink.
    float* outb = out + ENT_DIM + ncol;
    #pragma unroll
    for (int s = 0; s < SUBT; ++s) {
        const int base = s * 16 + 8 * h;            // rows base..base+7 for this lane-half
        const v4i t0 = *(const v4i*)&strig[base];
        const v4i t1 = *(const v4i*)&strig[base + 4];
        #pragma unroll
        for (int v = 0; v < 8; ++v) {
            const int tr = (v < 4) ? t0[v] : t1[v - 4];
            float* p = (tr >= 0) ? (outb + (size_t)tr * OUT_DIM) : sink;
            atomicAdd(p, acc[s][v]);
        }
    }
}

// ---------------------------------------------------------------------------
extern "C" void kernel_launch(void* const* d_in, const int* in_sizes, int n_in,
                              void* d_out, int out_size, void* d_ws, size_t ws_size,
                              hipStream_t stream)
{
    const float* ent       = (const float*)d_in[0];
    const float* rel       = (const float*)d_in[1];
    const float* rtab      = (const float*)d_in[2];
    const float* Win       = (const float*)d_in[3];
    const float* Wout      = (const float*)d_in[4];
    const int*   rtype_ids = (const int*)d_in[5];
    const int*   trig_ent  = (const int*)d_in[6];
    const int*   arg_trig  = (const int*)d_in[7];
    const int*   arg_relp  = (const int*)d_in[8];
    const int*   arg_entp  = (const int*)d_in[9];
    const int*   arg_is_in = (const int*)d_in[10];
    float*       out       = (float*)d_out;
    float*       sink      = (float*)d_ws;   // scratch sink for tail rows

    // 1) write trigger embeddings + zero the accumulation region
    const size_t total = (size_t)N_TRIG * OUT_DIM;
    ev_init_out<<<(unsigned)((total + 255) / 256), 256, 0, stream>>>(ent, trig_ent, out);

    // 2) masked GEMM + trigger segment-sum
    const int tiles = (N_ARGS + TILE_M - 1) / TILE_M;
    const size_t smem = (size_t)TILE_M * XS_STRIDE * sizeof(float)
                      + TILE_M * sizeof(float)
                      + TILE_M * sizeof(int);
    ev_args_wmma<<<tiles, 512, smem, stream>>>(ent, rel, rtab, Win, Wout,
                                               rtype_ids, arg_trig, arg_relp,
                                               arg_entp, arg_is_in, out, sink);
}